// GCNLayer_17703855194469
// MI455X (gfx1250) — compile-verified
//
#include <hip/hip_runtime.h>

// ---------------------------------------------------------------------------
// GCN layer for MI455X (gfx1250, wave32):
//   h   = segment_sum(x[src] * ew, dst)       (scatter w/ f32 atomics into L2)
//   out = h @ W^T + b                         (WMMA f32 16x16x4)
// ---------------------------------------------------------------------------

#define N_NODES 100000
#define N_EDGES 640000
#define D 128            // D_IN == D_OUT == 128

typedef float v8f __attribute__((ext_vector_type(8)));
typedef float v2f __attribute__((ext_vector_type(2)));

// ---------------- kernel 1: zero the accumulator h (workspace) -------------
__global__ void gcn_zero_kernel(float4* __restrict__ p, int n4) {
    int i = blockIdx.x * blockDim.x + threadIdx.x;
    if (i < n4) p[i] = make_float4(0.f, 0.f, 0.f, 0.f);
}

// ---------------- kernel 2: edge gather + weighted scatter-add -------------
// One wave32 per edge: each lane handles 4 contiguous features
// (global_load_b128 from x[src], 4x global_atomic_add_f32 into h[dst]).
__global__ void gcn_scatter_kernel(const float* __restrict__ x,
                                   const float* __restrict__ ew,
                                   const int* __restrict__ src,
                                   const int* __restrict__ dst,
                                   float* __restrict__ h) {
    const int wavesPerBlock = blockDim.x >> 5;
    const int e = blockIdx.x * wavesPerBlock + (threadIdx.x >> 5);
    if (e >= N_EDGES) return;
    const int lane = threadIdx.x & 31;

    const int   s = src[e];
    const int   d = dst[e];
    const float w = ew[e];

    const float4 v = *(const float4*)(x + (size_t)s * D + lane * 4);
    float* hp = h + (size_t)d * D + lane * 4;
    unsafeAtomicAdd(hp + 0, v.x * w);
    unsafeAtomicAdd(hp + 1, v.y * w);
    unsafeAtomicAdd(hp + 2, v.z * w);
    unsafeAtomicAdd(hp + 3, v.w * w);
}

// ---------------- kernel 3: out = h @ W^T + b  via V_WMMA_F32_16X16X4_F32 --
// Block = 256 threads = 8 waves. blockIdx.x selects the 16-row M tile,
// wave id selects the 16-col N tile (8 * 16 = 128 output cols).
//
// A (16x4 f32) lane layout: lanes 0-15 hold rows M=0..15 with K=0..1 in the
// two VGPRs; lanes 16-31 hold K=2..3  -> contiguous float2 of h per lane.
// B (4x16 f32) is the mirror with N across lanes; B[k][n] = W[n][k], so each
// lane's fragment is a contiguous float2 of W.
// C/D (16x16 f32, 8 VGPRs): VGPR r holds M=r (lanes 0-15) / M=r+8 (16-31).
__global__ void __launch_bounds__(256)
gcn_gemm_kernel(const float* __restrict__ h,
                const float* __restrict__ W,
                const float* __restrict__ b,
                float* __restrict__ out) {
    const int lane = threadIdx.x & 31;
    const int wave = threadIdx.x >> 5;      // N tile: 0..7
    const int row0 = blockIdx.x * 16;       // M tile base (100000 = 6250*16)
    const int col0 = wave * 16;
    const int l15  = lane & 15;
    const int rsel = lane >> 4;             // 0: K=0..1, 1: K=2..3

    const float* aptr = h + (size_t)(row0 + l15) * D + 2 * rsel;
    const float* bptr = W + (size_t)(col0 + l15) * D + 2 * rsel;

    v8f acc = {};
#pragma unroll
    for (int k = 0; k < D; k += 4) {
        v2f a  = *(const v2f*)(aptr + k);
        v2f bb = *(const v2f*)(bptr + k);
        acc = __builtin_amdgcn_wmma_f32_16x16x4_f32(
            /*neg_a=*/false, a, /*neg_b=*/false, bb,
            /*c_mod=*/(short)0, acc, /*reuse_a=*/false, /*reuse_b=*/false);
    }

    const float bias = b[col0 + l15];
    float* optr = out + (size_t)(row0 + 8 * rsel) * D + col0 + l15;
#pragma unroll
    for (int r = 0; r < 8; ++r) {
        optr[(size_t)r * D] = acc[r] + bias;
    }
}

// ---------------------------------------------------------------------------
extern "C" void kernel_launch(void* const* d_in, const int* in_sizes, int n_in,
                              void* d_out, int out_size, void* d_ws, size_t ws_size,
                              hipStream_t stream) {
    const float* x   = (const float*)d_in[0];   // [N_NODES, 128] f32
    const float* ew  = (const float*)d_in[1];   // [N_EDGES, 1]   f32
    const int*   src = (const int*)  d_in[2];   // [N_EDGES]      i32
    const int*   dst = (const int*)  d_in[3];   // [N_EDGES]      i32
    const float* Wm  = (const float*)d_in[4];   // [128, 128]     f32
    const float* bv  = (const float*)d_in[5];   // [128]          f32

    float* h   = (float*)d_ws;                  // [N_NODES, 128] accumulator
    float* out = (float*)d_out;                 // [N_NODES, 128]

    // 1) h = 0
    {
        const int n4 = (N_NODES * D) / 4;       // 3,200,000 float4
        const int threads = 256;
        const int blocks = (n4 + threads - 1) / threads;
        gcn_zero_kernel<<<blocks, threads, 0, stream>>>((float4*)h, n4);
    }

    // 2) scatter-add weighted messages (one wave per edge)
    {
        const int threads = 256;                // 8 waves / block
        const int wavesPerBlock = threads / 32;
        const int blocks = (N_EDGES + wavesPerBlock - 1) / wavesPerBlock;
        gcn_scatter_kernel<<<blocks, threads, 0, stream>>>(x, ew, src, dst, h);
    }

    // 3) out = h @ W^T + b  (WMMA)
    {
        const int threads = 256;                // 8 waves = 8 N tiles
        const int blocks = N_NODES / 16;        // 6250 M tiles
        gcn_gemm_kernel<<<blocks, threads, 0, stream>>>(h, Wm, bv, out);
    }
}